// FusedScaleMaskSoftmax_79903571575134
// MI455X (gfx1250) — compile-verified
//
#include <hip/hip_runtime.h>
#include <stdint.h>

typedef float v4f __attribute__((ext_vector_type(4)));

namespace {

constexpr float kScale    = 0.125f;     // 1/sqrt(64)
constexpr float kMaskFill = -10000.0f;

constexpr int SK  = 2048;               // softmax axis length
constexpr int SQ  = 2048;
constexpr int NH  = 16;
constexpr int ROW_BYTES  = SK * 4;      // 8192 B per data row
constexpr int MASK_BYTES = SK;          // 2048 B per mask row (numpy bool = 1 byte)
constexpr int WAVE = 32;                // wave32 on gfx1250
constexpr int WPB  = 4;                 // waves per block
constexpr int BLOCK = WPB * WAVE;       // 128 threads
constexpr int CHUNKS  = ROW_BYTES  / (WAVE * 16);  // 16 b128 chunks / data row
constexpr int MCHUNKS = MASK_BYTES / (WAVE * 16);  // 4  b128 chunks / mask row

__global__ __launch_bounds__(BLOCK) void fused_scale_mask_softmax(
    const float* __restrict__ inp,
    const unsigned char* __restrict__ mask,
    float* __restrict__ out,
    int nrows)
{
  // 4 waves * (8 KiB data + 2 KiB mask) = 40 KiB static LDS
  __shared__ __align__(16) unsigned char lds[WPB * (ROW_BYTES + MASK_BYTES)];

  const int w    = threadIdx.x >> 5;
  const int lane = threadIdx.x & (WAVE - 1);
  const int row  = blockIdx.x * WPB + w;
  if (row >= nrows) return;

  const int data_off = w * ROW_BYTES;
  const int mask_off = WPB * ROW_BYTES + w * MASK_BYTES;

  // ------- CDNA5 async DMA: stage data row + mask row into LDS -------------
  // ISA: each lane copies 16 B from global memory (VADDR) into LDS (VDST)
  // (same INST_OFFSET applied to both sides -> one base, 512B per wave-chunk)
  {
    // low 32 bits of a flat pointer to __shared__ == wave-relative LDS address
    const unsigned lds_d = (unsigned)(uintptr_t)(&lds[data_off]) + (unsigned)(lane * 16);
    const unsigned lds_m = (unsigned)(uintptr_t)(&lds[mask_off]) + (unsigned)(lane * 16);

    const unsigned long long g_d =
        (unsigned long long)(uintptr_t)(inp + (size_t)row * SK) + (unsigned)(lane * 16);

    const int ib   = row >> 15;          // row / (NH*SQ); NH*SQ = 32768
    const int iq   = row & (SQ - 1);
    const int mrow = (ib << 11) | iq;    // ib*SQ + iq  (mask broadcast over heads)
    const unsigned long long g_m =
        (unsigned long long)(uintptr_t)(mask + (size_t)mrow * SK) + (unsigned)(lane * 16);

    // data: single-use stream -> non-temporal (don't thrash L2)
#pragma unroll
    for (int k = 0; k < CHUNKS; ++k)
      asm volatile("global_load_async_to_lds_b128 %0, %1, off offset:%2 th:TH_LOAD_NT"
                   :: "v"(lds_d), "v"(g_d), "i"(k * 512) : "memory");
    // mask: reused by all 16 heads -> regular temporal (lives in 192MB L2)
#pragma unroll
    for (int k = 0; k < MCHUNKS; ++k)
      asm volatile("global_load_async_to_lds_b128 %0, %1, off offset:%2"
                   :: "v"(lds_m), "v"(g_m), "i"(k * 512) : "memory");

    asm volatile("s_wait_asynccnt 0" ::: "memory");
  }

  v4f* xrow = reinterpret_cast<v4f*>(&lds[data_off]);
  const unsigned* mraw = reinterpret_cast<const unsigned*>(&lds[mask_off]);

  // ------- pass 1: scale + mask, running max; write masked values back -----
  float mx = kMaskFill;
#pragma unroll
  for (int k = 0; k < CHUNKS; ++k) {
    const int i = k * WAVE + lane;
    v4f v = xrow[i];
    const unsigned mu = mraw[i];         // 4 mask bytes, little-endian
    v.x = (mu & 0x000000FFu) ? kMaskFill : v.x * kScale;
    v.y = (mu & 0x0000FF00u) ? kMaskFill : v.y * kScale;
    v.z = (mu & 0x00FF0000u) ? kMaskFill : v.z * kScale;
    v.w = (mu & 0xFF000000u) ? kMaskFill : v.w * kScale;
    mx = fmaxf(mx, fmaxf(fmaxf(v.x, v.y), fmaxf(v.z, v.w)));
    xrow[i] = v;
  }
#pragma unroll
  for (int d = WAVE / 2; d > 0; d >>= 1)
    mx = fmaxf(mx, __shfl_xor(mx, d, WAVE));

  // ------- pass 2: exp(x - max), sum; write exp back -----------------------
  float sum = 0.0f;
#pragma unroll
  for (int k = 0; k < CHUNKS; ++k) {
    const int i = k * WAVE + lane;
    v4f v = xrow[i];
    v.x = __expf(v.x - mx);
    v.y = __expf(v.y - mx);
    v.z = __expf(v.z - mx);
    v.w = __expf(v.w - mx);
    sum += (v.x + v.y) + (v.z + v.w);
    xrow[i] = v;
  }
#pragma unroll
  for (int d = WAVE / 2; d > 0; d >>= 1)
    sum += __shfl_xor(sum, d, WAVE);
  const float r = 1.0f / sum;

  // ------- pass 3: normalize, stream out with non-temporal b128 stores -----
  v4f* orow = reinterpret_cast<v4f*>(out + (size_t)row * SK);
#pragma unroll
  for (int k = 0; k < CHUNKS; ++k) {
    const int i = k * WAVE + lane;
    v4f v = xrow[i];
    v.x *= r; v.y *= r; v.z *= r; v.w *= r;
    __builtin_nontemporal_store(v, &orow[i]);
  }
}

} // namespace

extern "C" void kernel_launch(void* const* d_in, const int* in_sizes, int n_in,
                              void* d_out, int out_size, void* d_ws, size_t ws_size,
                              hipStream_t stream) {
  const float* inp = (const float*)d_in[0];
  const unsigned char* mask = (const unsigned char*)d_in[1];  // jnp bool -> 1 byte
  float* out = (float*)d_out;

  const int nrows = in_sizes[0] / SK;   // b*nh*sq = 65536
  dim3 grid((nrows + WPB - 1) / WPB);   // 16384 blocks
  dim3 block(BLOCK);                    // 128 threads = 4 wave32
  hipLaunchKernelGGL(fused_scale_mask_softmax, grid, block, 0, stream,
                     inp, mask, out, nrows);
}